// cross_scale_trans_49323404427936
// MI455X (gfx1250) — compile-verified
//
#include <hip/hip_runtime.h>

typedef float v8f __attribute__((ext_vector_type(8)));
typedef float v2f __attribute__((ext_vector_type(2)));

#define D_MODEL 128
#define D_CHL   64
#define D_FFN   256
#define MKEEP   16
#define DISTTH  4
#define NBS     132   // padded LDS row stride (128+4) to break bank conflicts

// ---------------------------------------------------------------------------
// CDNA5 async-copy helpers.
// GLOBAL_LOAD_ASYNC_TO_LDS_B32: per-lane LDS dest offset in VDST VGPR,
// 64-bit global address in VADDR pair, tracked by ASYNCcnt (ISA ch.10 / 15.18).
// Generic LDS pointers carry the LDS offset in their low 32 bits (ISA 10.2:
// LDS aperture).
// ---------------------------------------------------------------------------
__device__ __forceinline__ void async_copy_f32(float* lds_dst, const float* gsrc) {
    unsigned lofs = (unsigned)(unsigned long long)lds_dst;
    asm volatile("global_load_async_to_lds_b32 %0, %1, off"
                 :: "v"(lofs), "v"(gsrc) : "memory");
}

__device__ __forceinline__ void wait_async0() {
#if __has_builtin(__builtin_amdgcn_s_wait_asynccnt)
    __builtin_amdgcn_s_wait_asynccnt(0);
#else
    asm volatile("s_wait_asynccnt 0x0" ::: "memory");
#endif
}

// ---------------------------------------------------------------------------
// Kernel 1: pack coords to bytes + PE hidden = relu(vc @ pe_w1 + pe_b1)
// grid = N blocks of 64 threads
// ---------------------------------------------------------------------------
__global__ void pe_pack_kernel(const int* __restrict__ coords,
                               const float* __restrict__ w1,
                               const float* __restrict__ b1,
                               unsigned* __restrict__ packed,
                               float* __restrict__ hidden) {
    int n = blockIdx.x;
    int t = threadIdx.x;                 // 0..63
    int x = coords[n * 3 + 0];
    int y = coords[n * 3 + 1];
    int z = coords[n * 3 + 2];
    if (t == 0)
        packed[n] = (unsigned)x | ((unsigned)y << 8) | ((unsigned)z << 16);
    const float inv = 1.0f / 95.0f;      // / (GRID-1)
    float vx = (float)x * inv, vy = (float)y * inv, vz = (float)z * inv;
    float h = b1[t] + vx * w1[t] + vy * w1[64 + t] + vz * w1[128 + t];
    hidden[(size_t)n * 64 + t] = fmaxf(h, 0.0f);
}

// ---------------------------------------------------------------------------
// Kernel 2: O(N^2) Manhattan-distance stable top-16 (thread per point).
// Strict '<' insertion reproduces jax.lax.top_k tie-break (lower index first).
// ---------------------------------------------------------------------------
__global__ void topk_kernel(const unsigned* __restrict__ packed,
                            int* __restrict__ nbidx, int N) {
    int n = blockIdx.x * blockDim.x + threadIdx.x;
    if (n >= N) return;
    unsigned pc = packed[n];
    int x = (int)(pc & 255u), y = (int)((pc >> 8) & 255u), z = (int)((pc >> 16) & 255u);
    int dist[MKEEP], idxs[MKEEP];
#pragma unroll
    for (int i = 0; i < MKEEP; ++i) { dist[i] = 1 << 20; idxs[i] = -1; }
    int worst = 1 << 20;
    for (int j = 0; j < N; ++j) {
        unsigned q = packed[j];
        int dx = (int)(q & 255u) - x;          dx = dx < 0 ? -dx : dx;
        int dy = (int)((q >> 8) & 255u) - y;   dy = dy < 0 ? -dy : dy;
        int dz = (int)((q >> 16) & 255u) - z;  dz = dz < 0 ? -dz : dz;
        int d = dx + dy + dz;
        if (d <= DISTTH && d < worst) {
            int cd = d, cj = j;
#pragma unroll
            for (int p = 0; p < MKEEP; ++p) {
                if (cd < dist[p]) {
                    int td = dist[p], tj = idxs[p];
                    dist[p] = cd; idxs[p] = cj;
                    cd = td; cj = tj;
                }
            }
            worst = dist[MKEEP - 1];
        }
    }
#pragma unroll
    for (int i = 0; i < MKEEP; ++i)
        nbidx[(size_t)n * MKEEP + i] = (dist[i] <= DISTTH) ? idxs[i] : -1;
}

// ---------------------------------------------------------------------------
// Generic WMMA f32 GEMM: C = act(A[M,K] @ B[K,N] + bias + addsrc)
// block = (32,4): 4 waves, each owns one 16x16 output tile; all 4 share the
// same 16 A rows, so the 16xK A tile is async-staged into LDS ONCE per block
// (coalesced GLOBAL_LOAD_ASYNC_TO_LDS_B32), then read conflict-free
// (stride K+4 -> rows on distinct banks). grid = (Mrows/16, Ncols/64).
// A-layout: lanes 0-15 -> (m=lane, k=k0,k0+1); lanes 16-31 -> (m, k0+2,k0+3).
// C-layout: VGPR v -> row v (lanes<16) / v+8 (lanes>=16), col = lane&15.
// ---------------------------------------------------------------------------
__global__ void gemm16(const float* __restrict__ A, int lda,
                       const float* __restrict__ B, int ldb,
                       const float* __restrict__ bias,
                       const float* __restrict__ addsrc,
                       float* __restrict__ C, int ldc,
                       int K, int do_relu) {
    __shared__ float a_s[16 * (256 + 4)];   // max K = 256
    int row0 = blockIdx.x * 16;
    int col0 = (blockIdx.y * blockDim.y + threadIdx.y) * 16;
    int lane = threadIdx.x;                 // 0..31
    int t    = threadIdx.y * 32 + lane;     // 0..127
    int astr = K + 4;

    // cooperative async staging of the A tile
    for (int m = 0; m < 16; ++m) {
        const float* gp = A + (size_t)(row0 + m) * lda;
        for (int k = t; k < K; k += 128)
            async_copy_f32(&a_s[m * astr + k], gp + k);
    }
    wait_async0();
    __syncthreads();

    int mrow = lane & 15;
    int koff = (lane >> 4) << 1;            // 0 or 2
    v8f acc = {};
    for (int k0 = 0; k0 < K; k0 += 4) {
        v2f a, bb;
        a.x  = a_s[mrow * astr + k0 + koff];
        a.y  = a_s[mrow * astr + k0 + koff + 1];
        bb.x = B[(size_t)(k0 + koff) * ldb + col0 + mrow];
        bb.y = B[(size_t)(k0 + koff + 1) * ldb + col0 + mrow];
        acc = __builtin_amdgcn_wmma_f32_16x16x4_f32(false, a, false, bb,
                                                    (short)0, acc, false, false);
    }
    int col = col0 + mrow;
    float bv = bias ? bias[col] : 0.0f;
    int rbase = row0 + ((lane >> 4) << 3);
#pragma unroll
    for (int v = 0; v < 8; ++v) {
        int row = rbase + v;
        float val = acc[v] + bv;
        if (addsrc) val += addsrc[(size_t)row * ldc + col];
        if (do_relu) val = fmaxf(val, 0.0f);
        C[(size_t)row * ldc + col] = val;
    }
}

// ---------------------------------------------------------------------------
// Kernel 4: per-point attention. Block = 128 threads (4 waves), one point.
// Async-gathers neighbors to LDS, WMMA K/V projections from LDS, scalar Q,
// scores/softmax/out with the torch .view head-interleave remap:
//   k[n,h,m,d] = kproj[n, h*4 + m/4, (m%4)*32 + d]
// ---------------------------------------------------------------------------
__global__ void attn_kernel(const float* __restrict__ src,
                            const int* __restrict__ nbidx,
                            const float* __restrict__ q_w, const float* __restrict__ q_b,
                            const float* __restrict__ k_w, const float* __restrict__ k_b,
                            const float* __restrict__ v_w, const float* __restrict__ v_b,
                            float* __restrict__ hout) {
    __shared__ float nb_s[MKEEP * NBS];
    __shared__ float k_s[MKEEP * NBS];
    __shared__ float v_s[MKEEP * NBS];
    __shared__ float q_s[D_MODEL];
    __shared__ float sc_s[64];
    __shared__ int   idx_s[MKEEP];

    int n = blockIdx.x;
    int t = threadIdx.x;                 // 0..127

    // warm WGP cache for the shared weight panels (global_prefetch_b8)
    __builtin_prefetch(q_w + (size_t)t * D_MODEL, 0, 3);
    __builtin_prefetch(k_w + (size_t)t * D_MODEL, 0, 3);
    __builtin_prefetch(v_w + (size_t)t * D_MODEL, 0, 3);

    if (t < MKEEP) idx_s[t] = nbidx[(size_t)n * MKEEP + t];
    __syncthreads();

    // async gather of neighbor features into LDS; idx_s[m] is block-uniform,
    // so the async-vs-zero-fill branch is non-divergent.
#pragma unroll
    for (int m = 0; m < MKEEP; ++m) {
        int id = idx_s[m];
        if (id >= 0) {
            async_copy_f32(&nb_s[m * NBS + t], src + (size_t)id * D_MODEL + t);
        } else {
            nb_s[m * NBS + t] = 0.0f;    // reference zeroes invalid slots
        }
    }
    wait_async0();
    __syncthreads();

    // K/V projections: 16 tile jobs (8 col-tiles x {K,V}) across 4 waves
    int wave = t >> 5, lane = t & 31;
    int mrow = lane & 15;
    int koff = (lane >> 4) << 1;
    for (int job = wave; job < 16; job += 4) {
        int ct  = job >> 1;
        int isv = job & 1;
        const float* W  = isv ? v_w : k_w;
        const float* Bb = isv ? v_b : k_b;
        float* OUT      = isv ? v_s : k_s;
        int col0 = ct * 16;
        v8f acc = {};
        for (int k0 = 0; k0 < D_MODEL; k0 += 4) {
            v2f a, bb;
            a.x  = nb_s[mrow * NBS + k0 + koff];
            a.y  = nb_s[mrow * NBS + k0 + koff + 1];
            bb.x = W[(size_t)(k0 + koff) * D_MODEL + col0 + mrow];
            bb.y = W[(size_t)(k0 + koff + 1) * D_MODEL + col0 + mrow];
            acc = __builtin_amdgcn_wmma_f32_16x16x4_f32(false, a, false, bb,
                                                        (short)0, acc, false, false);
        }
        int col = col0 + mrow;
        float bias = Bb[col];            // bias applies even to zeroed rows
        int rbase = (lane >> 4) << 3;
#pragma unroll
        for (int v = 0; v < 8; ++v)
            OUT[(rbase + v) * NBS + col] = acc[v] + bias;
    }
    __syncthreads();

    // Q projection from neighbor 0 only
    float aq = q_b[t];
    for (int k = 0; k < D_MODEL; ++k)
        aq += nb_s[k] * q_w[(size_t)k * D_MODEL + t];
    q_s[t] = aq;
    __syncthreads();

    // scores with the interleaved .view remap; /sqrt(D_MODEL)
    if (t < 64) {
        int h = t >> 4, m = t & 15;
        const float* krow = &k_s[(h * 4 + (m >> 2)) * NBS + (m & 3) * 32];
        const float* qh   = &q_s[h * 32];
        float s = 0.0f;
#pragma unroll
        for (int d = 0; d < 32; ++d) s += qh[d] * krow[d];
        sc_s[t] = s * 0.08838834764831845f;   // 1/sqrt(128)
    }
    __syncthreads();

    // softmax over all 16 slots per head (invalid slots carry bias-only k/v,
    // exactly as the reference computes)
    if (t < 4) {
        float mx = -1e30f;
#pragma unroll
        for (int m = 0; m < MKEEP; ++m) mx = fmaxf(mx, sc_s[t * 16 + m]);
        float e[MKEEP], sum = 0.0f;
#pragma unroll
        for (int m = 0; m < MKEEP; ++m) { e[m] = expf(sc_s[t * 16 + m] - mx); sum += e[m]; }
        float inv = 1.0f / sum;
#pragma unroll
        for (int m = 0; m < MKEEP; ++m) sc_s[t * 16 + m] = e[m] * inv;
    }
    __syncthreads();

    // head_out[n, h*32+d] = sum_m attn[h][m] * v[n,h,m,d]
    int h = t >> 5, d = t & 31;
    float o = 0.0f;
#pragma unroll
    for (int m = 0; m < MKEEP; ++m)
        o += sc_s[h * 16 + m] * v_s[(h * 4 + (m >> 2)) * NBS + (m & 3) * 32 + d];
    hout[(size_t)n * D_MODEL + t] = o;
}

// ---------------------------------------------------------------------------
// LayerNorm over 128 channels, wave (32 lanes) per row, shfl reductions.
// ---------------------------------------------------------------------------
__global__ void ln_kernel(const float* __restrict__ x,
                          const float* __restrict__ g,
                          const float* __restrict__ b,
                          float* __restrict__ y) {
    int n = blockIdx.x, lane = threadIdx.x;
    const float* row = x + (size_t)n * D_MODEL;
    float v[4];
#pragma unroll
    for (int i = 0; i < 4; ++i) v[i] = row[lane + 32 * i];
    float s = v[0] + v[1] + v[2] + v[3];
#pragma unroll
    for (int o = 16; o > 0; o >>= 1) s += __shfl_xor(s, o, 32);
    float mu = s * (1.0f / 128.0f);
    float q = 0.0f;
#pragma unroll
    for (int i = 0; i < 4; ++i) { float d = v[i] - mu; q += d * d; }
#pragma unroll
    for (int o = 16; o > 0; o >>= 1) q += __shfl_xor(q, o, 32);
    float inv = rsqrtf(q * (1.0f / 128.0f) + 1e-5f);
    float* yr = y + (size_t)n * D_MODEL;
#pragma unroll
    for (int i = 0; i < 4; ++i) {
        int c = lane + 32 * i;
        yr[c] = (v[i] - mu) * inv * g[c] + b[c];
    }
}

// ---------------------------------------------------------------------------
// BatchNorm (train-mode) stats over N per channel; block per channel.
// ---------------------------------------------------------------------------
__global__ void bn_stats_kernel(const float* __restrict__ fused,
                                float* __restrict__ stats, int N) {
    int c = blockIdx.x, t = threadIdx.x;
    float s1 = 0.0f, s2 = 0.0f;
    for (int i = t; i < N; i += 256) {
        float v = fused[(size_t)i * D_CHL + c];
        s1 += v; s2 += v * v;
    }
    __shared__ float r1[256], r2[256];
    r1[t] = s1; r2[t] = s2;
    __syncthreads();
    for (int o = 128; o > 0; o >>= 1) {
        if (t < o) { r1[t] += r1[t + o]; r2[t] += r2[t + o]; }
        __syncthreads();
    }
    if (t == 0) {
        float mu = r1[0] / (float)N;
        stats[c] = mu;
        stats[64 + c] = r2[0] / (float)N - mu * mu;
    }
}

__global__ void bn_apply_kernel(const float* __restrict__ fused,
                                const float* __restrict__ stats,
                                const float* __restrict__ g,
                                const float* __restrict__ b,
                                float* __restrict__ out, int total) {
    int i = blockIdx.x * 256 + threadIdx.x;
    if (i >= total) return;
    int c = i & 63;
    float v = (fused[i] - stats[c]) * rsqrtf(stats[64 + c] + 1e-3f) * g[c] + b[c];
    out[i] = fmaxf(v, 0.0f);
}

// ---------------------------------------------------------------------------
extern "C" void kernel_launch(void* const* d_in, const int* in_sizes, int n_in,
                              void* d_out, int out_size, void* d_ws, size_t ws_size,
                              hipStream_t stream) {
    const float* features = (const float*)d_in[0];
    const int*   coords   = (const int*)  d_in[1];
    const float* pe_w1 = (const float*)d_in[2];  const float* pe_b1 = (const float*)d_in[3];
    const float* pe_w2 = (const float*)d_in[4];  const float* pe_b2 = (const float*)d_in[5];
    const float* in_w  = (const float*)d_in[6];  const float* in_b  = (const float*)d_in[7];
    const float* q_w   = (const float*)d_in[8];  const float* q_b   = (const float*)d_in[9];
    const float* k_w   = (const float*)d_in[10]; const float* k_b   = (const float*)d_in[11];
    const float* v_w   = (const float*)d_in[12]; const float* v_b   = (const float*)d_in[13];
    const float* o_w   = (const float*)d_in[14]; const float* o_b   = (const float*)d_in[15];
    const float* n1_g  = (const float*)d_in[16]; const float* n1_b  = (const float*)d_in[17];
    const float* l1_w  = (const float*)d_in[18]; const float* l1_b  = (const float*)d_in[19];
    const float* l2_w  = (const float*)d_in[20]; const float* l2_b  = (const float*)d_in[21];
    const float* n3_g  = (const float*)d_in[22]; const float* n3_b  = (const float*)d_in[23];
    const float* fu_w  = (const float*)d_in[24]; const float* fu_b  = (const float*)d_in[25];
    const float* bn_g  = (const float*)d_in[26]; const float* bn_b  = (const float*)d_in[27];

    const int N = in_sizes[0] / D_CHL;   // 8192

    // workspace carve-up (all buffers fully written before read each call)
    char* ws = (char*)d_ws;
    size_t off = 0;
    auto carve = [&](size_t bytes) -> char* {
        char* p = ws + off;
        off = (off + bytes + 255) & ~(size_t)255;
        return p;
    };
    unsigned* packed = (unsigned*)carve((size_t)N * 4);
    int*      nbidx  = (int*)     carve((size_t)N * MKEEP * 4);
    float*    hidden = (float*)   carve((size_t)N * 64 * 4);
    float*    src    = (float*)   carve((size_t)N * D_MODEL * 4);
    float*    hout   = (float*)   carve((size_t)N * D_MODEL * 4);
    float*    res1   = (float*)   carve((size_t)N * D_MODEL * 4);
    float*    tgt    = (float*)   carve((size_t)N * D_MODEL * 4);
    float*    ffh    = (float*)   carve((size_t)N * D_FFN * 4);
    float*    res2   = (float*)   carve((size_t)N * D_MODEL * 4);
    float*    tgt2   = (float*)   carve((size_t)N * D_MODEL * 4);
    float*    fused  = (float*)   carve((size_t)N * D_CHL * 4);
    float*    stats  = (float*)   carve(128 * 4);
    (void)ws_size; (void)n_in; (void)out_size;

    dim3 gblk(32, 4);

    // 1) pack coords + PE hidden
    pe_pack_kernel<<<N, 64, 0, stream>>>(coords, pe_w1, pe_b1, packed, hidden);

    // 2) neighbor top-16
    topk_kernel<<<(N + 255) / 256, 256, 0, stream>>>(packed, nbidx, N);

    // 3) src = features@in_w + in_b  +  hidden@pe_w2 + pe_b2   (split-K, WMMA)
    gemm16<<<dim3(N / 16, 2), gblk, 0, stream>>>(features, 64, in_w, 128,
                                                 in_b, nullptr, src, 128, 64, 0);
    gemm16<<<dim3(N / 16, 2), gblk, 0, stream>>>(hidden, 64, pe_w2, 128,
                                                 pe_b2, src, src, 128, 64, 0);

    // 4) per-point attention (async gather + WMMA K/V projections in LDS)
    attn_kernel<<<N, 128, 0, stream>>>(src, nbidx, q_w, q_b, k_w, k_b, v_w, v_b, hout);

    // 5) out proj + residual, LN1
    gemm16<<<dim3(N / 16, 2), gblk, 0, stream>>>(hout, 128, o_w, 128,
                                                 o_b, src, res1, 128, 128, 0);
    ln_kernel<<<N, 32, 0, stream>>>(res1, n1_g, n1_b, tgt);

    // 6) FFN: relu(tgt@l1_w+l1_b) @ l2_w + l2_b + tgt, LN2
    gemm16<<<dim3(N / 16, 4), gblk, 0, stream>>>(tgt, 128, l1_w, 256,
                                                 l1_b, nullptr, ffh, 256, 128, 1);
    gemm16<<<dim3(N / 16, 2), gblk, 0, stream>>>(ffh, 256, l2_w, 128,
                                                 l2_b, tgt, res2, 128, 256, 0);
    ln_kernel<<<N, 32, 0, stream>>>(res2, n3_g, n3_b, tgt2);

    // 7) fusion: [features | tgt2] @ fu_w + fu_b   (split-K, WMMA)
    gemm16<<<dim3(N / 16, 1), gblk, 0, stream>>>(features, 64, fu_w, 64,
                                                 fu_b, nullptr, fused, 64, 64, 0);
    gemm16<<<dim3(N / 16, 1), gblk, 0, stream>>>(tgt2, 128, fu_w + 64 * 64, 64,
                                                 nullptr, fused, fused, 64, 128, 0);

    // 8) BatchNorm (train stats) + ReLU
    bn_stats_kernel<<<64, 256, 0, stream>>>(fused, stats, N);
    bn_apply_kernel<<<(N * D_CHL + 255) / 256, 256, 0, stream>>>(
        fused, stats, bn_g, bn_b, (float*)d_out, N * D_CHL);
}